// SimpleFusionModule_50672024158819
// MI455X (gfx1250) — compile-verified
//
#include <hip/hip_runtime.h>

typedef __attribute__((ext_vector_type(2))) float v2f;
typedef __attribute__((ext_vector_type(4))) float v4f;
typedef __attribute__((ext_vector_type(8))) float v8f;

#define STYLES 18
#define MROWS 64          // batch
#define KBLK 32           // K block per LDS stage
#define NTILE 128         // N columns per workgroup
#define SA 36             // padded LDS k-stride for A  (bank-conflict-free: 36 = 4*9)
#define SBT 36            // padded LDS k-stride for Bt (transposed [n][k])

// ---------------------------------------------------------------------------
// Concat: X0[s][b][0:1024] = { x1[b][s][0:512], x2[b][s][0:512] }
// x1,x2 are [B=64, S=18, 512]; X0 is style-major [S, B, 1024].
// ---------------------------------------------------------------------------
__global__ __launch_bounds__(256) void concat_kernel(const float* __restrict__ x1,
                                                     const float* __restrict__ x2,
                                                     float* __restrict__ X0) {
    int i = blockIdx.x * 256 + threadIdx.x;
    int k = i & 1023;
    int b = (i >> 10) & 63;
    int s = i >> 16;                      // 64*1024 = 2^16
    size_t src = (size_t)b * (STYLES * 512) + (size_t)s * 512;
    float v = (k < 512) ? x1[src + k] : x2[src + (k - 512)];
    X0[i] = v;
}

// ---------------------------------------------------------------------------
// Grouped GEMM over styles:  Y[s] = act( X[s] * W[s] + bias[s] )
// grid = (N/128, S); block = 256 threads (8 waves).
// Wave w: rows m0 = (w>>1)*16, cols nb = (w&1)*64 (4 WMMA 16x16 tiles).
// Double-buffered LDS, register-prefetch software pipeline.
// A in LDS as [m][k] (stride 36);  B in LDS transposed as [n][k] (stride 36)
// so both WMMA fragments are single aligned ds_load_b64s.
// ---------------------------------------------------------------------------
__global__ __launch_bounds__(256) void grouped_gemm_wmma(
    const float* __restrict__ X, const float* __restrict__ W,
    const float* __restrict__ bias, float* __restrict__ Y,
    int K, int N,
    long in_ss, int in_rs,
    long out_ss, int out_rs,
    int relu)
{
    __shared__ float As[2][MROWS * SA];    // 2 x  9216 B
    __shared__ float Bt[2][NTILE * SBT];   // 2 x 18432 B   (total 55296 B)

    const int s      = blockIdx.y;
    const int ntile0 = blockIdx.x * NTILE;

    const int tid  = threadIdx.x;
    const int lane = tid & 31;
    const int wave = tid >> 5;
    const int ml   = lane & 15;        // M index in tile (N index for B/C)
    const int kh   = lane >> 4;        // lane-half: k pair base 2*kh, C rows +8
    const int m0   = (wave >> 1) * 16;
    const int nb   = (wave & 1) * 64;

    const float* Ap = X + (size_t)s * in_ss;
    const float* Wp = W + (size_t)s * (size_t)K * (size_t)N + ntile0;

    // --- staging decomposition -----------------------------------------
    // A: thread loads rows a_m and a_m+32, 4 consecutive k (b128)
    const int a_m  = tid >> 3;          // 0..31
    const int a_k4 = (tid & 7) * 4;
    // B: thread owns one column n, 16 consecutive k values (per-lane b32
    // global loads -> transposed b64 LDS stores, conflict-free both ways)
    const int b_n  = tid & 127;
    const int b_k0 = (tid >> 7) * 16;   // 0 or 16

    const float* aptr0 = Ap + (size_t)a_m * in_rs + a_k4;
    const float* aptr1 = Ap + (size_t)(a_m + 32) * in_rs + a_k4;
    const float* bptr  = Wp + (size_t)b_k0 * N + b_n;

    v4f  ra0, ra1;
    float rb[16];

    v8f acc[4];
    acc[0] = (v8f)0.0f; acc[1] = (v8f)0.0f; acc[2] = (v8f)0.0f; acc[3] = (v8f)0.0f;

    auto fetch = [&](int kb) {
        ra0 = *(const v4f*)(aptr0 + kb);
        ra1 = *(const v4f*)(aptr1 + kb);
        const float* bp = bptr + (size_t)kb * N;
        #pragma unroll
        for (int q = 0; q < 16; ++q)
            rb[q] = __builtin_nontemporal_load(bp + (size_t)q * N);  // weights: stream once
    };

    auto stage = [&](int buf) {
        *(v4f*)(&As[buf][a_m * SA + a_k4])        = ra0;
        *(v4f*)(&As[buf][(a_m + 32) * SA + a_k4]) = ra1;
        #pragma unroll
        for (int q = 0; q < 8; ++q) {
            v2f p; p.x = rb[2 * q]; p.y = rb[2 * q + 1];
            *(v2f*)(&Bt[buf][b_n * SBT + b_k0 + 2 * q]) = p;   // transposed store
        }
    };

    auto compute = [&](int buf) {
        #pragma unroll
        for (int kk = 0; kk < KBLK; kk += 4) {
            const int kf = kk + 2 * kh;
            v2f a = *(const v2f*)(&As[buf][(m0 + ml) * SA + kf]);
            #pragma unroll
            for (int j = 0; j < 4; ++j) {
                v2f b = *(const v2f*)(&Bt[buf][(nb + 16 * j + ml) * SBT + kf]);
                acc[j] = __builtin_amdgcn_wmma_f32_16x16x4_f32(
                    false, a, false, b, (short)0, acc[j], false, false);
            }
        }
    };

    // --- software pipeline ----------------------------------------------
    fetch(0);
    stage(0);
    __syncthreads();

    const int nblk = K / KBLK;
    for (int i = 0; i < nblk; ++i) {
        if (i + 1 < nblk) fetch((i + 1) * KBLK);   // overlap HBM with compute
        compute(i & 1);
        if (i + 1 < nblk) {
            stage((i + 1) & 1);                    // other buffer: no race
            __syncthreads();
        }
    }

    // --- epilogue: bias + optional ReLU ----------------------------------
    #pragma unroll
    for (int j = 0; j < 4; ++j) {
        const int n   = ntile0 + nb + 16 * j + ml;
        const float bv = bias[(size_t)s * N + n];
        #pragma unroll
        for (int v = 0; v < 8; ++v) {
            const int m = m0 + v + 8 * kh;         // C/D layout: M = v + 8*half
            float val = acc[j][v] + bv;
            if (relu) val = fmaxf(val, 0.0f);
            Y[(size_t)s * out_ss + (size_t)m * out_rs + n] = val;
        }
    }
}

// ---------------------------------------------------------------------------
extern "C" void kernel_launch(void* const* d_in, const int* in_sizes, int n_in,
                              void* d_out, int out_size, void* d_ws, size_t ws_size,
                              hipStream_t stream) {
    const float* x1 = (const float*)d_in[0];
    const float* x2 = (const float*)d_in[1];
    const float* W1 = (const float*)d_in[2];  const float* b1 = (const float*)d_in[3];
    const float* W2 = (const float*)d_in[4];  const float* b2 = (const float*)d_in[5];
    const float* W3 = (const float*)d_in[6];  const float* b3 = (const float*)d_in[7];
    const float* W4 = (const float*)d_in[8];  const float* b4 = (const float*)d_in[9];
    const float* W5 = (const float*)d_in[10]; const float* b5 = (const float*)d_in[11];
    float* out = (float*)d_out;

    // ping-pong activation buffers in workspace (style-major [S, 64, dim])
    float* buf0 = (float*)d_ws;                                     // up to 18.9 MB
    float* buf1 = (float*)((char*)d_ws + (size_t)20 * 1024 * 1024); // up to  9.5 MB

    dim3 blk(256);

    concat_kernel<<<dim3((STYLES * 64 * 1024) / 256), blk, 0, stream>>>(x1, x2, buf0);

    // L1: 1024 -> 2048, ReLU
    grouped_gemm_wmma<<<dim3(2048 / NTILE, STYLES), blk, 0, stream>>>(
        buf0, W1, b1, buf1, 1024, 2048,
        (long)64 * 1024, 1024, (long)64 * 2048, 2048, 1);
    // L2: 2048 -> 4096, ReLU
    grouped_gemm_wmma<<<dim3(4096 / NTILE, STYLES), blk, 0, stream>>>(
        buf1, W2, b2, buf0, 2048, 4096,
        (long)64 * 2048, 2048, (long)64 * 4096, 4096, 1);
    // L3: 4096 -> 2048, ReLU
    grouped_gemm_wmma<<<dim3(2048 / NTILE, STYLES), blk, 0, stream>>>(
        buf0, W3, b3, buf1, 4096, 2048,
        (long)64 * 4096, 4096, (long)64 * 2048, 2048, 1);
    // L4: 2048 -> 1024, ReLU
    grouped_gemm_wmma<<<dim3(1024 / NTILE, STYLES), blk, 0, stream>>>(
        buf1, W4, b4, buf0, 2048, 1024,
        (long)64 * 2048, 2048, (long)64 * 1024, 1024, 1);
    // L5: 1024 -> 512, no ReLU, write [B, S, 512] layout directly
    grouped_gemm_wmma<<<dim3(512 / NTILE, STYLES), blk, 0, stream>>>(
        buf0, W5, b5, out, 1024, 512,
        (long)64 * 1024, 1024, (long)512, STYLES * 512, 0);
}